// Recycler_29901562314767
// MI455X (gfx1250) — compile-verified
//
#include <hip/hip_runtime.h>
#include <hip/hip_bf16.h>
#include <math.h>

// Problem constants (match reference)
#define BATCH 8
#define N_TOK 2048
#define D_DIM 256
#define KC    16   // K-chunk staged in LDS per iteration
#define LDSR  20   // padded LDS row stride (floats): conflict-free & 16B-aligned

typedef __attribute__((ext_vector_type(2))) float v2f;
typedef __attribute__((ext_vector_type(8))) float v8f;

__device__ __forceinline__ v8f wmma_f32(v2f a, v2f b, v8f c) {
    // D = A(16x4) * B(4x16) + C, full fp32 matrix op on CDNA5
    return __builtin_amdgcn_wmma_f32_16x16x4_f32(
        /*neg_a=*/false, a, /*neg_b=*/false, b,
        /*c_mod=*/(short)0, c, /*reuse_a=*/false, /*reuse_b=*/false);
}

// Async-copy a 128x16 fp32 tile (global rows stride D_DIM) into padded LDS.
// Uses CDNA5 GLOBAL_LOAD_ASYNC_TO_LDS_B128 (ASYNCcnt-tracked, bypasses VGPRs).
__device__ __forceinline__ void async_fill(const float* __restrict__ src,
                                           float* lds_base, int tid) {
#pragma unroll
    for (int i = 0; i < 2; ++i) {
        int idx = tid + i * 256;      // 512 float4s total (128 rows x 4 float4)
        int row = idx >> 2;
        int c4  = idx & 3;
        unsigned long long gaddr =
            (unsigned long long)(uintptr_t)(src + (size_t)row * D_DIM + c4 * 4);
        unsigned ldsoff =
            (unsigned)(uintptr_t)(lds_base + row * LDSR + c4 * 4);
        asm volatile("global_load_async_to_lds_b128 %0, %1, off"
                     :: "v"(ldsoff), "v"(gaddr) : "memory");
    }
}

__device__ __forceinline__ void wait_async_and_sync() {
    asm volatile("s_wait_asynccnt 0x0" ::: "memory");
    __syncthreads();
}

// out[m][e] = sum_d X[m][d] * W[e][d] + bias[e]
// grid: (N/128, M/128), block: 256 (8 waves, 4x2 wave tiling of 128x128)
__global__ __launch_bounds__(256) void proj_kernel(const float* __restrict__ X,
                                                   const float* __restrict__ W,
                                                   const float* __restrict__ bias,
                                                   float* __restrict__ out) {
    __shared__ alignas(16) float As[2][128 * LDSR];
    __shared__ alignas(16) float Bs[2][128 * LDSR];

    const int tid  = threadIdx.x;
    const int lane = tid & 31, wave = tid >> 5;
    const int wm = wave & 3, wn = wave >> 2;
    const int m = lane & 15, half = lane >> 4;
    const int kb = 2 * half;
    const int tileN = blockIdx.x * 128;
    const int tileM = blockIdx.y * 128;

    const float* Xg = X + (size_t)tileM * D_DIM;
    const float* Wg = W + (size_t)tileN * D_DIM;   // B[k][n] = W[n][k]

    v8f acc[2][4];
    v8f zero = {};
#pragma unroll
    for (int s = 0; s < 2; ++s)
#pragma unroll
        for (int t = 0; t < 4; ++t) acc[s][t] = zero;

    async_fill(Xg, As[0], tid);
    async_fill(Wg, Bs[0], tid);

    int buf = 0;
    for (int kc = 0; kc < D_DIM; kc += KC) {
        wait_async_and_sync();             // buffer[buf] resident for all waves
        int nxt = buf ^ 1;
        if (kc + KC < D_DIM) {             // stream next chunk while computing
            async_fill(Xg + kc + KC, As[nxt], tid);
            async_fill(Wg + kc + KC, Bs[nxt], tid);
        }
        const float* Asb = As[buf];
        const float* Bsb = Bs[buf];
#pragma unroll
        for (int k0 = 0; k0 < KC; k0 += 4) {
            v2f a0 = *(const v2f*)&Asb[(wm * 32 +      m) * LDSR + k0 + kb];
            v2f a1 = *(const v2f*)&Asb[(wm * 32 + 16 + m) * LDSR + k0 + kb];
#pragma unroll
            for (int t = 0; t < 4; ++t) {
                v2f b = *(const v2f*)&Bsb[(wn * 64 + t * 16 + m) * LDSR + k0 + kb];
                acc[0][t] = wmma_f32(a0, b, acc[0][t]);
                acc[1][t] = wmma_f32(a1, b, acc[1][t]);
            }
        }
        buf = nxt;
    }

#pragma unroll
    for (int s = 0; s < 2; ++s)
#pragma unroll
        for (int t = 0; t < 4; ++t) {
            int col = tileN + wn * 64 + t * 16 + m;
            float bv = bias[col];
#pragma unroll
            for (int v = 0; v < 8; ++v) {
                int row = tileM + wm * 32 + s * 16 + v + 8 * half;
                out[(size_t)row * D_DIM + col] = acc[s][t][v] + bv;
            }
        }
}

// scores[b][i][j] = mask[i][j] ? 10*tanh(Q[b][i]·K[b][j]) : -inf
// grid: (N_TOK/128, N_TOK/128, BATCH), block: 256
__global__ __launch_bounds__(256) void scores_kernel(const float* __restrict__ Q,
                                                     const float* __restrict__ Km,
                                                     const int* __restrict__ mask,
                                                     float* __restrict__ out) {
    __shared__ alignas(16) float Qs[2][128 * LDSR];
    __shared__ alignas(16) float Ks[2][128 * LDSR];

    const int tid  = threadIdx.x;
    const int lane = tid & 31, wave = tid >> 5;
    const int wm = wave & 3, wn = wave >> 2;
    const int m = lane & 15, half = lane >> 4;
    const int kb = 2 * half;
    const int tileN = blockIdx.x * 128;
    const int tileM = blockIdx.y * 128;
    const int b     = blockIdx.z;

    const float* Qg = Q  + ((size_t)b * N_TOK + tileM) * D_DIM;
    const float* Kg = Km + ((size_t)b * N_TOK + tileN) * D_DIM;  // B[k][j]=K[j][k]

    v8f acc[2][4];
    v8f zero = {};
#pragma unroll
    for (int s = 0; s < 2; ++s)
#pragma unroll
        for (int t = 0; t < 4; ++t) acc[s][t] = zero;

    async_fill(Qg, Qs[0], tid);
    async_fill(Kg, Ks[0], tid);

    int buf = 0;
    for (int kc = 0; kc < D_DIM; kc += KC) {
        wait_async_and_sync();
        int nxt = buf ^ 1;
        if (kc + KC < D_DIM) {
            async_fill(Qg + kc + KC, Qs[nxt], tid);
            async_fill(Kg + kc + KC, Ks[nxt], tid);
        }
        const float* Qsb = Qs[buf];
        const float* Ksb = Ks[buf];
#pragma unroll
        for (int k0 = 0; k0 < KC; k0 += 4) {
            v2f a0 = *(const v2f*)&Qsb[(wm * 32 +      m) * LDSR + k0 + kb];
            v2f a1 = *(const v2f*)&Qsb[(wm * 32 + 16 + m) * LDSR + k0 + kb];
#pragma unroll
            for (int t = 0; t < 4; ++t) {
                v2f b2 = *(const v2f*)&Ksb[(wn * 64 + t * 16 + m) * LDSR + k0 + kb];
                acc[0][t] = wmma_f32(a0, b2, acc[0][t]);
                acc[1][t] = wmma_f32(a1, b2, acc[1][t]);
            }
        }
        buf = nxt;
    }

    const float ninf = -__builtin_inff();
#pragma unroll
    for (int s = 0; s < 2; ++s)
#pragma unroll
        for (int t = 0; t < 4; ++t) {
            int j = tileN + wn * 64 + t * 16 + m;
#pragma unroll
            for (int v = 0; v < 8; ++v) {
                int i = tileM + wm * 32 + s * 16 + v + 8 * half;
                float sv = 10.0f * tanhf(acc[s][t][v]);
                float r = (mask[(size_t)i * N_TOK + j] == 0) ? ninf : sv;
                out[((size_t)b * N_TOK + i) * N_TOK + j] = r;
            }
        }
}

// picked_logit[r] = x[r]·Wl + bl ; one wave (32 lanes) per row
__global__ __launch_bounds__(256) void logit_kernel(const float* __restrict__ X,
                                                    const float* __restrict__ Wl,
                                                    const float* __restrict__ bl,
                                                    float* __restrict__ out) {
    int lane = threadIdx.x & 31;
    int wave = threadIdx.x >> 5;
    int row  = blockIdx.x * 8 + wave;
    const float* xr = X + (size_t)row * D_DIM;
    float s = 0.0f;
#pragma unroll
    for (int i = 0; i < D_DIM; i += 32) s += xr[i + lane] * Wl[i + lane];
#pragma unroll
    for (int off = 16; off > 0; off >>= 1) s += __shfl_down(s, off, 32);
    if (lane == 0) out[row] = s + bl[0];
}

extern "C" void kernel_launch(void* const* d_in, const int* in_sizes, int n_in,
                              void* d_out, int out_size, void* d_ws, size_t ws_size,
                              hipStream_t stream) {
    const float* x    = (const float*)d_in[0];
    const int*   mask = (const int*)d_in[1];
    const float* Wq   = (const float*)d_in[2];
    const float* bq   = (const float*)d_in[3];
    const float* Wk   = (const float*)d_in[4];
    const float* bk   = (const float*)d_in[5];
    const float* Wl   = (const float*)d_in[6];
    const float* bl   = (const float*)d_in[7];

    // Workspace: Q then K, each B*N*D fp32 (16 MB each, 32 MB total)
    float* Qws = (float*)d_ws;
    float* Kws = Qws + (size_t)BATCH * N_TOK * D_DIM;

    // d_out: picked_logit (B*N) first, then scores (B*N*N)
    float* logit_out  = (float*)d_out;
    float* scores_out = logit_out + (size_t)BATCH * N_TOK;

    dim3 gProj(D_DIM / 128, (BATCH * N_TOK) / 128);   // (2, 128)
    proj_kernel<<<gProj, 256, 0, stream>>>(x, Wq, bq, Qws);
    proj_kernel<<<gProj, 256, 0, stream>>>(x, Wk, bk, Kws);
    logit_kernel<<<(BATCH * N_TOK) / 8, 256, 0, stream>>>(x, Wl, bl, logit_out);
    scores_kernel<<<dim3(N_TOK / 128, N_TOK / 128, BATCH), 256, 0, stream>>>(
        Qws, Kws, mask, scores_out);
}